// ArcEmbedding_40870908788984
// MI455X (gfx1250) — compile-verified
//
#include <hip/hip_runtime.h>
#include <hip/hip_bf16.h>

// Problem constants (match reference)
#define B_      32
#define S_      4096
#define H_      768
#define HALF_   384
#define V_      64
#define THREADS 1024
#define ROWS    256                 // rows per block; 512 blocks total
#define TABLE_FLOATS (V_ * H_)      // 49152
#define TABLE_BYTES  (TABLE_FLOATS * 4)  // 196608 B -> fits CDNA5 320KB/WGP LDS

// -ln(10000)/128
#define NEG_LN_BASE_OVER_D (-0.07195578415606394f)

// ---------------------------------------------------------------------------
// Kernel 1: per-batch max over S of coords.x / coords.y, clamped to >= 1.0
// ---------------------------------------------------------------------------
__global__ __launch_bounds__(256) void axis_max_kernel(
    const int* __restrict__ coords, float* __restrict__ axmax) {
  __shared__ int sx[256];
  __shared__ int sy[256];
  const int b = blockIdx.x;
  const int t = threadIdx.x;
  int mx = 0, my = 0;
  const int* cb = coords + (size_t)b * S_ * 2;
  for (int i = t; i < S_; i += 256) {
    mx = max(mx, cb[2 * i + 0]);
    my = max(my, cb[2 * i + 1]);
  }
  sx[t] = mx; sy[t] = my;
  __syncthreads();
  for (int off = 128; off > 0; off >>= 1) {
    if (t < off) {
      sx[t] = max(sx[t], sx[t + off]);
      sy[t] = max(sy[t], sy[t + off]);
    }
    __syncthreads();
  }
  if (t == 0) {
    axmax[2 * b + 0] = fmaxf((float)sx[0], 1.0f);
    axmax[2 * b + 1] = fmaxf((float)sy[0], 1.0f);
  }
}

// ---------------------------------------------------------------------------
// Kernel 2: stage full 192KB emb_table into LDS via CDNA5 async DMA, then
// gather + 3D-RoPE rotate, non-temporal streaming stores.
// ---------------------------------------------------------------------------
__global__ __launch_bounds__(THREADS) void rope_gather_kernel(
    const int*   __restrict__ ids,
    const int*   __restrict__ coords,
    const float* __restrict__ table,
    const float* __restrict__ axmax,
    float*       __restrict__ out) {
  extern __shared__ float lds_tab[];  // TABLE_BYTES dynamic LDS

  const int t    = threadIdx.x;
  const int lane = t & 31;
  const int wave = t >> 5;

  // ---- Stage emb_table -> LDS with async-to-LDS DMA (ASYNCcnt tracked) ----
  {
    // Low 32 bits of a generic pointer to LDS == workgroup-relative LDS byte
    // offset (LDS aperture: addr[31:0] is the LDS address).
    unsigned lds_base = (unsigned)(size_t)(&lds_tab[0]);
    const char* gbase = (const char*)table;
    // 12288 x 16B chunks, 12 per thread
    for (int i = t; i < TABLE_FLOATS / 4; i += THREADS) {
      unsigned lds_addr = lds_base + (unsigned)i * 16u;
      const char* gaddr = gbase + (size_t)i * 16u;
      asm volatile("global_load_async_to_lds_b128 %0, %1, off"
                   :: "v"(lds_addr), "v"(gaddr)
                   : "memory");
    }
    asm volatile("s_wait_asynccnt 0" ::: "memory");
  }
  __syncthreads();  // all waves' DMA complete before any wave gathers

  // ---- Per-lane inverse frequencies: f = lane + 32*m, m = 0..3 ----
  float invf4[4];
#pragma unroll
  for (int m = 0; m < 4; ++m) {
    invf4[m] = __expf((float)(lane + 32 * m) * NEG_LN_BASE_OVER_D);
  }

  const int row0 = blockIdx.x * ROWS;   // 256 rows; never straddles a batch
  const int b    = row0 / S_;
  const float xmax = axmax[2 * b + 0];
  const float ymax = axmax[2 * b + 1];
  const float smax = (float)(S_ - 1);   // 4095 >= 1, no clamp needed

  // 32 waves * 8 rows = 256 rows per block
#pragma unroll 1
  for (int i = 0; i < ROWS / (THREADS / 32); ++i) {
    const int lr  = wave * (ROWS / (THREADS / 32)) + i;
    const int row = row0 + lr;
    const int s   = row - b * S_;

    const int   id = ids[row];                    // wave-uniform -> scalar load
    const float cx = (float)coords[2 * row + 0] / xmax;
    const float cy = (float)coords[2 * row + 1] / ymax;
    const float cs = (float)s / smax;

    const float* trow = &lds_tab[id * H_];
    float*       orow = out + (size_t)row * H_;

#pragma unroll
    for (int k = 0; k < 12; ++k) {
      const int   j     = lane + 32 * k;          // 0..383
      // axis = j/128 = k/4 (compile-time per unrolled k); f = lane + 32*(k&3)
      const float coord = (k < 4) ? cx : ((k < 8) ? cy : cs);
      const float ang   = coord * invf4[k & 3];
      float sn, cn;
      __sincosf(ang, &sn, &cn);
      const float e1 = trow[j];                   // ds_load, conflict-free
      const float e2 = trow[j + HALF_];
      // out[j] = e1*cos - e2*sin ; out[j+384] = e2*cos + e1*sin
      __builtin_nontemporal_store(fmaf(e1, cn, -e2 * sn), orow + j);
      __builtin_nontemporal_store(fmaf(e2, cn,  e1 * sn), orow + j + HALF_);
    }
  }
}

// ---------------------------------------------------------------------------
extern "C" void kernel_launch(void* const* d_in, const int* in_sizes, int n_in,
                              void* d_out, int out_size, void* d_ws, size_t ws_size,
                              hipStream_t stream) {
  const int*   ids    = (const int*)d_in[0];    // input_ids [B,S]
  const int*   coords = (const int*)d_in[1];    // coords    [B,S,2]
  const float* table  = (const float*)d_in[2];  // emb_table [V,H]
  float*       out    = (float*)d_out;          // [B,S,H] fp32
  float*       axmax  = (float*)d_ws;           // 2 floats per batch

  axis_max_kernel<<<B_, 256, 0, stream>>>(coords, axmax);
  rope_gather_kernel<<<(B_ * S_) / ROWS, THREADS, TABLE_BYTES, stream>>>(
      ids, coords, table, axmax, out);
}